// Denoise_15049565405686
// MI455X (gfx1250) — compile-verified
//
#include <hip/hip_runtime.h>
#include <hip/hip_bf16.h>

// ---------------------------------------------------------------------------
// Bilateral 5x5 filter (sigma_spatial=1.5, sigma_color=0.1, reflect padding)
// on [B=2, C=3, H=1024, W=1024] fp32.  Memory-light (~50MB total), trans-pipe
// bound (25 v_exp_f32 per pixel).  Uses CDNA5 async global->LDS staging.
// ---------------------------------------------------------------------------

#define AS1 __attribute__((address_space(1)))
#define AS3 __attribute__((address_space(3)))

namespace {
constexpr int TILE = 32;
constexpr int HALO = 2;              // kernel 5x5 -> pad 2
constexpr int KS   = 5;
constexpr int PADW = TILE + 2 * HALO;   // 36
constexpr int LSTR = PADW + 1;          // 37 floats: conflict-free stride
constexpr int IMG_H = 1024;
constexpr int IMG_W = 1024;
constexpr int NTHREADS = 256;           // 8 wave32 per WGP-resident block
}

__device__ __forceinline__ int reflect_idx(int i, int n) {
    i = (i < 0) ? -i : i;                 // reflect low side (no edge repeat)
    i = (i >= n) ? (2 * n - 2 - i) : i;   // reflect high side
    return i;
}

__device__ __forceinline__ float fast_exp2(float x) {
#if __has_builtin(__builtin_amdgcn_exp2f)
    return __builtin_amdgcn_exp2f(x);     // v_exp_f32
#else
    return exp2f(x);
#endif
}

__global__ __launch_bounds__(NTHREADS)
void bilateral5x5_kernel(const float* __restrict__ img,
                         float* __restrict__ out) {
    __shared__ float tile[PADW * LSTR];   // 36*37*4 = 5328 B of the 320 KB LDS

    const int plane = blockIdx.z;                       // b*c plane, 0..5
    const long long pbase = (long long)plane * (long long)(IMG_H * IMG_W);
    const int ty0 = blockIdx.y * TILE;
    const int tx0 = blockIdx.x * TILE;
    const int tid = threadIdx.x;

    // ---- Stage 36x36 halo tile into LDS (reflect borders) ----
    // Async per-lane global->LDS copies: no VGPR round-trip, ASYNCcnt tracked.
    #pragma unroll
    for (int base = 0; base < PADW * PADW; base += NTHREADS) {
        const int idx = base + tid;
        if (idx < PADW * PADW) {
            const int r = idx / PADW;
            const int c = idx - r * PADW;
            const int gy = reflect_idx(ty0 + r - HALO, IMG_H);
            const int gx = reflect_idx(tx0 + c - HALO, IMG_W);
            const float* gp = img + pbase + (long long)gy * IMG_W + gx;
#if __has_builtin(__builtin_amdgcn_global_load_async_to_lds_b32)
            __builtin_amdgcn_global_load_async_to_lds_b32(
                (AS1 int*)(void*)gp,
                (AS3 int*)(void*)&tile[r * LSTR + c],
                /*imm offset=*/0, /*cpol=*/0);
#else
            tile[r * LSTR + c] = *gp;     // fallback: vmem load + ds_store
#endif
        }
    }
#if __has_builtin(__builtin_amdgcn_global_load_async_to_lds_b32)
#if __has_builtin(__builtin_amdgcn_s_wait_asynccnt)
    __builtin_amdgcn_s_wait_asynccnt(0);
#else
    asm volatile("s_wait_asynccnt 0" ::: "memory");
#endif
#endif
    __syncthreads();

    // ---- Spatial Gaussian 1D taps (sigma=1.5), normalized; products fold ----
    const float k1[KS] = {0.12007838f, 0.23388076f, 0.29208153f,
                          0.23388076f, 0.12007838f};
    // exp(-0.5*(d/0.1)^2) = exp(-50*d^2) = exp2(-72.134752*d^2)
    constexpr float COLOR_SCALE = -72.134752f;

    const int px  = tid & (TILE - 1);     // 0..31 : column (coalesced lanes)
    const int py0 = tid >> 5;             // 0..7  : row group

    #pragma unroll
    for (int s = 0; s < 4; ++s) {
        const int py = py0 + 8 * s;       // 0..31
        const float* trow = &tile[py * LSTR + px];
        const float cen = trow[HALO * LSTR + HALO];

        float wsum = 1e-8f;               // reference adds EPS to weight sum
        float psum = 0.0f;
        #pragma unroll
        for (int i = 0; i < KS; ++i) {
            #pragma unroll
            for (int j = 0; j < KS; ++j) {
                const float p = trow[i * LSTR + j];
                const float d = p - cen;
                const float w = fast_exp2(d * d * COLOR_SCALE) * (k1[i] * k1[j]);
                wsum += w;
                psum = __builtin_fmaf(w, p, psum);
            }
        }

#if __has_builtin(__builtin_amdgcn_rcpf)
        const float r = __builtin_amdgcn_rcpf(wsum);   // v_rcp_f32 (~1 ulp)
#else
        const float r = 1.0f / wsum;
#endif
        out[pbase + (long long)(ty0 + py) * IMG_W + (tx0 + px)] = psum * r;
    }
}

extern "C" void kernel_launch(void* const* d_in, const int* in_sizes, int n_in,
                              void* d_out, int out_size, void* d_ws, size_t ws_size,
                              hipStream_t stream) {
    (void)n_in; (void)d_ws; (void)ws_size; (void)out_size;
    const float* img = (const float*)d_in[0];
    float* out = (float*)d_out;

    const int planes = in_sizes[0] / (IMG_H * IMG_W);   // B*C = 6
    dim3 grid(IMG_W / TILE, IMG_H / TILE, planes);      // 32 x 32 x 6
    dim3 block(NTHREADS);
    bilateral5x5_kernel<<<grid, block, 0, stream>>>(img, out);
}